// MoEGate_50946902065259
// MI455X (gfx1250) — compile-verified
//
#include <hip/hip_runtime.h>

#define TOP_K       8
#define N_EXPERTS   64
#define N_GROUP     8
#define TOPK_GROUP  4
#define ROUTED_SCALE 1.0f
#define AUX_ALPHA   0.001f

#define DIM            2048
#define TOKENS_PER_WG  128
#define TOKENS_PER_WAVE 16
#define KCHUNK         128
#define WLDS_STRIDE    132   // 128+4 ; 132 % 64 == 4 -> conflict-free b64 B-frag loads
#define SLDS_STRIDE    67    // odd  -> conflict-free row & column access
#define SMEM_FLOATS    (TOKENS_PER_WG * SLDS_STRIDE)   // 8576 floats (> 64*132=8448)

typedef __attribute__((ext_vector_type(2))) float v2f;
typedef __attribute__((ext_vector_type(8))) float v8f;

__global__ __launch_bounds__(256)
void moe_gate_main(const float* __restrict__ hidden,   // [T, DIM]
                   const float* __restrict__ weight,   // [64, DIM]
                   float* __restrict__ out,            // [T*8 idx | T*8 w | 1 aux]
                   float* __restrict__ ws,             // [nb*64 sumscore | nb*64 count]
                   int T, int seq_len, int n_batch)
{
    __shared__ __align__(16) float smem[SMEM_FLOATS];
    __shared__ int cnt[N_EXPERTS];

    const int tid  = threadIdx.x;
    const int lane = tid & 31;
    const int wave = tid >> 5;
    const int tok0 = blockIdx.x * TOKENS_PER_WG;

    // ---------------- GEMM phase: logits = hidden @ weight^T via fp32 WMMA ----
    const int half = lane >> 4;      // 0: K=k,k+1   1: K=k+2,k+3  (A 16x4 layout)
    const int col  = lane & 15;      // M row (A) / N col (B)

    v8f acc[4];
#pragma unroll
    for (int t = 0; t < 4; ++t)
#pragma unroll
        for (int r = 0; r < 8; ++r) acc[t][r] = 0.0f;

    const int mytok = tok0 + wave * TOKENS_PER_WAVE + col;
    const float* __restrict__ arow = hidden + (size_t)mytok * DIM + 2 * half;

    for (int kc = 0; kc < DIM; kc += KCHUNK) {
        // cooperative stage of weight chunk [64][128] into LDS (float4, coalesced)
#pragma unroll
        for (int i = 0; i < (N_EXPERTS * KCHUNK / 4) / 256; ++i) {
            int idx = tid + i * 256;          // 0..2047 float4 slots
            int row = idx >> 5;               // 32 float4 per row
            int c4  = idx & 31;
            float4 v = *(const float4*)(weight + (size_t)row * DIM + kc + c4 * 4);
            *(float4*)&smem[row * WLDS_STRIDE + c4 * 4] = v;
        }
        __syncthreads();

#pragma unroll 4
        for (int k = 0; k < KCHUNK; k += 4) {
            v2f a = *(const v2f*)(arow + kc + k);   // A frag: 16x4 fp32 tile
#pragma unroll
            for (int t = 0; t < 4; ++t) {           // 4 N-tiles -> 64 experts
                v2f b = *(const v2f*)&smem[(16 * t + col) * WLDS_STRIDE + k + 2 * half];
                acc[t] = __builtin_amdgcn_wmma_f32_16x16x4_f32(
                    false, a, false, b, (short)0, acc[t], false, false);
            }
        }
        __syncthreads();
    }

    // ---------------- spill logits to LDS: slds[token_local][expert] ----------
    // C/D layout: lane l, vgpr r -> M = r + 8*(l>=16), N = l&15
#pragma unroll
    for (int t = 0; t < 4; ++t)
#pragma unroll
        for (int r = 0; r < 8; ++r)
            smem[(wave * 16 + r + 8 * half) * SLDS_STRIDE + 16 * t + col] = acc[t][r];
    if (tid < N_EXPERTS) cnt[tid] = 0;
    __syncthreads();

    // ---------------- per-token routing (one thread per token) ----------------
    if (tid < TOKENS_PER_WG) {
        const int tokL = tid;
        const int tokG = tok0 + tokL;
        float s[N_EXPERTS];
        const float* rowp = &smem[tokL * SLDS_STRIDE];
#pragma unroll
        for (int e = 0; e < N_EXPERTS; ++e) s[e] = rowp[e];

        // softmax over 64 experts
        float m = s[0];
#pragma unroll
        for (int e = 1; e < N_EXPERTS; ++e) m = fmaxf(m, s[e]);
        float sum = 0.0f;
#pragma unroll
        for (int e = 0; e < N_EXPERTS; ++e) { s[e] = expf(s[e] - m); sum += s[e]; }
        const float inv = 1.0f / sum;
        float* roww = &smem[tokL * SLDS_STRIDE];
#pragma unroll
        for (int e = 0; e < N_EXPERTS; ++e) { s[e] *= inv; roww[e] = s[e]; }

        // group scores = max over each group of 8
        float g[N_GROUP];
#pragma unroll
        for (int gi = 0; gi < N_GROUP; ++gi) {
            float gm = s[gi * 8];
#pragma unroll
            for (int j = 1; j < 8; ++j) gm = fmaxf(gm, s[gi * 8 + j]);
            g[gi] = gm;
        }
        // keep top TOPK_GROUP groups (lowest index wins ties, like lax.top_k)
        unsigned keep = 0;
#pragma unroll
        for (int r = 0; r < TOPK_GROUP; ++r) {
            float best = -1.0f; int bi = 0;
#pragma unroll
            for (int gi = 0; gi < N_GROUP; ++gi)
                if (!((keep >> gi) & 1u) && g[gi] > best) { best = g[gi]; bi = gi; }
            keep |= 1u << bi;
        }
        // top-8 experts among kept groups
        unsigned long long picked = 0ull;
        int idxs[TOP_K]; float wts[TOP_K]; float wsum = 0.0f;
#pragma unroll
        for (int j = 0; j < TOP_K; ++j) {
            float best = -1.0f; int bi = 0;
#pragma unroll
            for (int e = 0; e < N_EXPERTS; ++e) {
                bool ok = ((keep >> (e >> 3)) & 1u) && !((picked >> e) & 1ull);
                if (ok && s[e] > best) { best = s[e]; bi = e; }
            }
            picked |= 1ull << bi;
            idxs[j] = bi; wts[j] = best; wsum += best;
        }
        const float scale = ROUTED_SCALE / fmaxf(wsum, 1e-9f);
#pragma unroll
        for (int j = 0; j < TOP_K; ++j) {
            out[(size_t)tokG * TOP_K + j] = (float)idxs[j];
            out[(size_t)T * TOP_K + (size_t)tokG * TOP_K + j] = wts[j] * scale;
            atomicAdd(&cnt[idxs[j]], 1);
        }
    }
    __syncthreads();

    // ---------------- aux-loss partials: per-batch score sums & pick counts ---
    if (tid < N_EXPERTS) {
        float colsum = 0.0f;
#pragma unroll 4
        for (int t = 0; t < TOKENS_PER_WG; ++t) colsum += smem[t * SLDS_STRIDE + tid];
        const int batch = tok0 / seq_len;   // 128 | 4096 -> whole WG in one batch
        atomicAdd(&ws[batch * N_EXPERTS + tid], colsum);
        atomicAdd(&ws[(n_batch + batch) * N_EXPERTS + tid], (float)cnt[tid]);
    }
}

__global__ __launch_bounds__(256)
void moe_gate_aux(const float* __restrict__ ws, float* __restrict__ out,
                  int n_batch, int seq_len, unsigned long long aux_off)
{
    __shared__ float red[256];
    const int tid = threadIdx.x;
    const int n = n_batch * N_EXPERTS;
    float v = 0.0f;
    if (tid < n) v = ws[tid] * ws[n + tid];   // sumscore * count
    red[tid] = v;
    __syncthreads();
    for (int s = 128; s > 0; s >>= 1) {
        if (tid < s) red[tid] += red[tid + s];
        __syncthreads();
    }
    if (tid == 0) {
        // alpha * E / (B * S^2 * K)
        float factor = AUX_ALPHA * (float)N_EXPERTS /
                       ((float)n_batch * (float)seq_len * (float)seq_len * (float)TOP_K);
        out[aux_off] = red[0] * factor;
    }
}

extern "C" void kernel_launch(void* const* d_in, const int* in_sizes, int n_in,
                              void* d_out, int out_size, void* d_ws, size_t ws_size,
                              hipStream_t stream)
{
    const float* hidden = (const float*)d_in[0];   // [4,4096,2048] f32
    const float* weight = (const float*)d_in[1];   // [64,2048] f32
    float* out = (float*)d_out;
    float* ws  = (float*)d_ws;

    const int T = in_sizes[0] / DIM;     // 16384
    const int n_batch = 4;
    const int seq_len = T / n_batch;     // 4096

    hipMemsetAsync(d_ws, 0, (size_t)(2 * n_batch * N_EXPERTS) * sizeof(float), stream);

    dim3 grid(T / TOKENS_PER_WG), block(256);
    moe_gate_main<<<grid, block, 0, stream>>>(hidden, weight, out, ws, T, seq_len, n_batch);
    moe_gate_aux<<<1, 256, 0, stream>>>(ws, out, n_batch, seq_len,
                                        (unsigned long long)2 * T * TOP_K);
}